// WeightedGfbModule_79517024518806
// MI455X (gfx1250) — compile-verified
//
#include <hip/hip_runtime.h>
#include <hip/hip_bf16.h>

typedef __attribute__((ext_vector_type(2))) float v2f;
typedef __attribute__((ext_vector_type(4))) float v4f;
typedef __attribute__((ext_vector_type(8))) float v8f;

#define NB 256      // graphs
#define NC 2048     // feature channels
#define NH 512      // hidden
#define NN 32768    // nodes

// ---------------------------------------------------------------------------
// fp32 WMMA GEMM: C[M,N] = A[M,K] * B + bias(optional, per-column)
//   TRANSB=0: B is [K,N] row-major (B[k*N + n])
//   TRANSB=1: B is [N,K] row-major (B[n*K + k])  -> computes A * B^T
// One wave computes one 16x16 tile; 4 waves per block cover 4 N-tiles.
// v_wmma_f32_16x16x4_f32: native fp32 => same precision class as the fp32
// reference; these GEMMs are tiny (~1.1 GFLOP total) and fully L2-resident.
// ---------------------------------------------------------------------------
template <bool TRANSB>
__global__ void wmma_gemm_f32(const float* __restrict__ A,
                              const float* __restrict__ Bm,
                              const float* __restrict__ bias,
                              float* __restrict__ Cm,
                              int M, int Nn, int K) {
    const int lane = threadIdx.x & 31;
    const int wave = threadIdx.x >> 5;
    const int tile_n = blockIdx.x * 4 + wave;   // N/16 tiles
    const int tile_m = blockIdx.y;              // M/16 tiles
    const int half = lane >> 4;                 // 0: K=0,1 | 1: K=2,3
    const int l    = lane & 15;
    const int m0 = tile_m * 16;
    const int n0 = tile_n * 16;

    v8f acc = {};
    const float* arow = A + (size_t)(m0 + l) * K;
    const float* btrow = TRANSB ? (Bm + (size_t)(n0 + l) * K) : nullptr;

    for (int k0 = 0; k0 < K; k0 += 4) {
        const int ka = k0 + 2 * half;
        v2f a, b;
        a.x = arow[ka];
        a.y = arow[ka + 1];
        if (TRANSB) {
            b.x = btrow[ka];
            b.y = btrow[ka + 1];
        } else {
            b.x = Bm[(size_t)ka * Nn + n0 + l];
            b.y = Bm[(size_t)(ka + 1) * Nn + n0 + l];
        }
        // (neg_a, A, neg_b, B, c_mod, C, reuse_a, reuse_b)
        acc = __builtin_amdgcn_wmma_f32_16x16x4_f32(
            false, a, false, b, (short)0, acc, false, false);
    }

    const float bs = bias ? bias[n0 + l] : 0.0f;
#pragma unroll
    for (int r = 0; r < 8; ++r) {
        // D layout: VGPR r, lanes 0-15 -> row m0+r ; lanes 16-31 -> row m0+8+r
        Cm[(size_t)(m0 + r + 8 * half) * Nn + (n0 + l)] = acc[r] + bs;
    }
}

// r[g] = dot(q[g,:], bv)  (tiny: 256x512)
__global__ void qdotb_kernel(const float* __restrict__ q,
                             const float* __restrict__ bv,
                             float* __restrict__ r) {
    int g = threadIdx.x;
    if (g < NB) {
        float s = 0.0f;
        for (int h = 0; h < NH; ++h) s += q[(size_t)g * NH + h] * bv[h];
        r[g] = s;
    }
}

// w[n] = (feats[n,:] . P[seg[n],:] + r[seg[n]]) / NH   — feats pass 1.
// feats streamed non-temporally (no reuse, 256MB > L2); P stays L2-hot.
__global__ void score_kernel(const float* __restrict__ feats,
                             const float* __restrict__ P,
                             const float* __restrict__ r,
                             const int* __restrict__ seg,
                             float* __restrict__ w) {
    __shared__ float red[256];
    const int n = blockIdx.x;
    const int t = threadIdx.x;
    const int g = seg[n];
    const v4f* f4 = (const v4f*)(feats + (size_t)n * NC);
    const v4f* p4 = (const v4f*)(P + (size_t)g * NC);
    float s = 0.0f;
#pragma unroll
    for (int i = t; i < NC / 4; i += 256) {
        v4f a = __builtin_nontemporal_load(f4 + i);  // th:NT stream
        v4f b = p4[i];                               // L2-resident
        s += a.x * b.x + a.y * b.y + a.z * b.z + a.w * b.w;
    }
    red[t] = s;
    __syncthreads();
    for (int st = 128; st > 0; st >>= 1) {
        if (t < st) red[t] += red[t + st];
        __syncthreads();
    }
    if (t == 0) w[n] = (red[0] + r[g]) * (1.0f / (float)NH);
}

__device__ __forceinline__ int lower_bound_seg(const int* seg, int target) {
    int lo = 0, hi = NN;
    while (lo < hi) {
        int mid = (lo + hi) >> 1;
        if (seg[mid] < target) lo = mid + 1; else hi = mid;
    }
    return lo;
}

// Per-graph softmax over node scores: wn[n] = exp(w[n]-max)/sum
__global__ void segsoftmax_kernel(const int* __restrict__ seg,
                                  const float* __restrict__ w,
                                  float* __restrict__ wn) {
    __shared__ float red[256];
    __shared__ int se[2];
    const int g = blockIdx.x;
    const int t = threadIdx.x;
    if (t < 2) se[t] = lower_bound_seg(seg, g + t);
    __syncthreads();
    const int s0 = se[0], e0 = se[1];

    float m = -INFINITY;
    for (int n = s0 + t; n < e0; n += 256) m = fmaxf(m, w[n]);
    red[t] = m;
    __syncthreads();
    for (int st = 128; st > 0; st >>= 1) {
        if (t < st) red[t] = fmaxf(red[t], red[t + st]);
        __syncthreads();
    }
    m = red[0];
    __syncthreads();

    float sum = 0.0f;
    for (int n = s0 + t; n < e0; n += 256) sum += __expf(w[n] - m);
    red[t] = sum;
    __syncthreads();
    for (int st = 128; st > 0; st >>= 1) {
        if (t < st) red[t] += red[t + st];
        __syncthreads();
    }
    const float inv = 1.0f / red[0];
    for (int n = s0 + t; n < e0; n += 256) wn[n] = __expf(w[n] - m) * inv;
}

// h[g, c] = sum_{n in graph g} wn[n] * feats[n, c]  — feats pass 2.
// grid (B, NC/1024); each thread owns 4 channels (global_load_b128 per lane,
// 512B contiguous per wave per node). feats streamed non-temporally.
__global__ void aggregate_kernel(const int* __restrict__ seg,
                                 const float* __restrict__ wn,
                                 const float* __restrict__ feats,
                                 float* __restrict__ out) {
    __shared__ int se[2];
    const int g = blockIdx.x;
    const int t = threadIdx.x;
    const int c = blockIdx.y * 1024 + t * 4;
    if (t < 2) se[t] = lower_bound_seg(seg, g + t);
    __syncthreads();
    const int s0 = se[0], e0 = se[1];

    v4f acc = {};
#pragma unroll 2
    for (int n = s0; n < e0; ++n) {
        const float cf = wn[n];  // uniform broadcast, L2-resident
        const v4f f = __builtin_nontemporal_load(
            (const v4f*)(feats + (size_t)n * NC + c));
        acc.x += cf * f.x;
        acc.y += cf * f.y;
        acc.z += cf * f.z;
        acc.w += cf * f.w;
    }
    v4f* o = (v4f*)(out + (size_t)g * (2 * NC) + NC + c);
    __builtin_nontemporal_store(acc, o);   // written once, never re-read
}

// out[g, 0:NC] = sfb[g, :]   (float4 vectorized copy into strided output rows)
__global__ void copy_sfb_kernel(const float* __restrict__ sfb,
                                float* __restrict__ out) {
    const int idx = blockIdx.x * blockDim.x + threadIdx.x; // over B*NC/4
    const int g  = idx / (NC / 4);
    const int c4 = idx % (NC / 4);
    const v4f v = ((const v4f*)sfb)[idx];
    __builtin_nontemporal_store(v, (v4f*)(out + (size_t)g * (2 * NC)) + c4);
}

extern "C" void kernel_launch(void* const* d_in, const int* in_sizes, int n_in,
                              void* d_out, int out_size, void* d_ws, size_t ws_size,
                              hipStream_t stream) {
    const float* sfb   = (const float*)d_in[0]; // [B, C]
    const float* feats = (const float*)d_in[1]; // [N, C]
    const float* Wq    = (const float*)d_in[2]; // [C, H]
    const float* bq    = (const float*)d_in[3]; // [H]
    const float* Wv    = (const float*)d_in[4]; // [C, H]
    const float* bv    = (const float*)d_in[5]; // [H]
    const int*   seg   = (const int*)d_in[6];   // [N] sorted
    float* out = (float*)d_out;                 // [B, 2C]

    char* ws = (char*)d_ws;
    float* q  = (float*)ws; ws += (size_t)NB * NH * sizeof(float);  // 512 KB
    float* P  = (float*)ws; ws += (size_t)NB * NC * sizeof(float);  // 2 MB
    float* w  = (float*)ws; ws += (size_t)NN * sizeof(float);       // 128 KB
    float* wn = (float*)ws; ws += (size_t)NN * sizeof(float);       // 128 KB
    float* r  = (float*)ws; ws += (size_t)NB * sizeof(float);       // 1 KB

    // q = sfb @ Wq + bq   [256,512] : M=256, N=512, K=2048
    wmma_gemm_f32<false><<<dim3(NH / 64, NB / 16), 128, 0, stream>>>(
        sfb, Wq, bq, q, NB, NH, NC);

    // r[g] = q[g,:] . bv
    qdotb_kernel<<<1, 256, 0, stream>>>(q, bv, r);

    // P = q @ Wv^T        [256,2048] : M=256, N=2048, K=512
    wmma_gemm_f32<true><<<dim3(NC / 64, NB / 16), 128, 0, stream>>>(
        q, Wv, nullptr, P, NB, NC, NH);

    // w[n] = (feats[n] . P[seg[n]] + r[seg[n]]) / H    (feats pass 1)
    score_kernel<<<NN, 256, 0, stream>>>(feats, P, r, seg, w);

    // per-graph softmax -> wn
    segsoftmax_kernel<<<NB, 256, 0, stream>>>(seg, w, wn);

    // out[:, C:2C] = segment_sum(wn * feats)           (feats pass 2)
    aggregate_kernel<<<dim3(NB, NC / 1024), 256, 0, stream>>>(seg, wn, feats, out);

    // out[:, 0:C] = sfb
    copy_sfb_kernel<<<(NB * NC / 4) / 256, 256, 0, stream>>>(sfb, out);
}